// ConvOffset2d_31404800868787
// MI455X (gfx1250) — compile-verified
//
#include <hip/hip_runtime.h>

typedef __attribute__((ext_vector_type(2))) float v2f;
typedef __attribute__((ext_vector_type(8))) float v8f;

#define BATCH   8
#define CIN     64
#define HDIM    128
#define WDIM    128
#define HW      (HDIM * WDIM)        // 16384
#define OCH     64
#define KPTS    9                    // 3x3
#define CKTOT   (CIN * KPTS)         // 576
#define POS_T   64                   // spatial positions per block tile
#define CCH     8                    // channels per K-chunk
#define KC      (CCH * KPTS)         // 72 (multiple of 4 for WMMA K-steps)
#define KCP     (KC + 2)             // padded row stride (even -> 8B aligned pairs)
#define NTILES  (HW / POS_T)         // 256 tiles per batch image

__global__ __launch_bounds__(256)
void dcn_fused_wmma_f32(const float* __restrict__ x,
                        const float* __restrict__ offs,
                        const float* __restrict__ weight,
                        const float* __restrict__ bias,
                        float* __restrict__ out) {
  // ---- LDS: 9216*2 + 18944*2 = 56320 bytes ----
  __shared__ int   sI0[KPTS * POS_T], sI1[KPTS * POS_T];
  __shared__ int   sI2[KPTS * POS_T], sI3[KPTS * POS_T];
  __shared__ float sB0[KPTS * POS_T], sB1[KPTS * POS_T];
  __shared__ float sB2[KPTS * POS_T], sB3[KPTS * POS_T];
  __shared__ __align__(16) float wS[OCH * KCP];      // [oc][ck] slab
  __shared__ __align__(16) float colS[POS_T * KCP];  // [pos][ck] slab

  const int tid  = threadIdx.x;
  const int b    = blockIdx.x / NTILES;
  const int tile = blockIdx.x % NTILES;
  const int pos0 = tile * POS_T;

  // ---------- Stage A: per-(k,pos) bilinear indices & weights ----------
  for (int e = tid; e < KPTS * POS_T; e += 256) {
    const int k  = e / POS_T;
    const int p  = e % POS_T;
    const int gp = pos0 + p;
    const int pyi = gp >> 7;       // / WDIM
    const int pxi = gp & 127;      // % WDIM
    const int ky = k / 3, kx = k % 3;

    const float oy = offs[((size_t)b * (2 * KPTS) + 2 * k    ) * HW + gp];
    const float ox = offs[((size_t)b * (2 * KPTS) + 2 * k + 1) * HW + gp];
    const float py = oy + (float)(pyi - 1 + ky);   // stride 1, pad 1, dil 1
    const float px = ox + (float)(pxi - 1 + kx);

    const float fy = floorf(py), fx = floorf(px);
    const float ly = py - fy,    lx = px - fx;
    const int y0 = (int)fy, x0 = (int)fx;
    const int y1 = y0 + 1,  x1 = x0 + 1;

    const float vy0 = (y0 >= 0 && y0 < HDIM) ? 1.0f : 0.0f;
    const float vy1 = (y1 >= 0 && y1 < HDIM) ? 1.0f : 0.0f;
    const float vx0 = (x0 >= 0 && x0 < WDIM) ? 1.0f : 0.0f;
    const float vx1 = (x1 >= 0 && x1 < WDIM) ? 1.0f : 0.0f;

    const int cy0 = min(max(y0, 0), HDIM - 1);
    const int cy1 = min(max(y1, 0), HDIM - 1);
    const int cx0 = min(max(x0, 0), WDIM - 1);
    const int cx1 = min(max(x1, 0), WDIM - 1);

    sI0[e] = cy0 * WDIM + cx0;  sB0[e] = (1.0f - ly) * (1.0f - lx) * vy0 * vx0;
    sI1[e] = cy0 * WDIM + cx1;  sB1[e] = (1.0f - ly) * lx          * vy0 * vx1;
    sI2[e] = cy1 * WDIM + cx0;  sB2[e] = ly * (1.0f - lx)          * vy1 * vx0;
    sI3[e] = cy1 * WDIM + cx1;  sB3[e] = ly * lx                   * vy1 * vx1;
  }

  // ---------- wave / fragment mapping ----------
  const int lane   = tid & 31;
  const int wave   = tid >> 5;
  const int hi     = lane >> 4;                 // upper half-wave selector
  const int l16    = lane & 15;
  const int ocRow  = (wave & 3) * 16 + l16;     // A-fragment row (M)
  const int posA   = ((wave >> 2) * 2) * 16 + l16;  // tile-relative N, block 0
  const int posB   = posA + 16;                      // block 1

  v8f acc0 = {}, acc1 = {};

  // ---------- Stage B/C: channel-chunk loop ----------
  for (int c0 = 0; c0 < CIN; c0 += CCH) {
    __syncthreads();   // LDS slabs free from previous chunk's WMMA reads

    // weight slab: wS[oc][j], j = (c-c0)*9+k  -> contiguous in global per oc row
    const float* wg = weight + (size_t)c0 * KPTS;
    for (int t = tid; t < OCH * KC; t += 256) {
      const int oc = t / KC;
      const int j  = t - oc * KC;
      wS[oc * KCP + j] = wg[(size_t)oc * CKTOT + j];
    }
    // prefetch next chunk's weight rows into cache while we compute
    if (c0 + CCH < CIN && tid < OCH) {
      __builtin_prefetch(weight + (size_t)tid * CKTOT + (c0 + CCH) * KPTS, 0, 0);
    }

    // col slab: bilinear gather using precomputed coeffs
    for (int t = tid; t < POS_T * KC; t += 256) {
      const int p  = t / KC;
      const int j  = t - p * KC;
      const int cr = j / KPTS;
      const int k  = j - cr * KPTS;
      const int sk = k * POS_T + p;
      const float* xc = x + ((size_t)b * CIN + (c0 + cr)) * HW;
      const float v = sB0[sk] * xc[sI0[sk]] + sB1[sk] * xc[sI1[sk]]
                    + sB2[sk] * xc[sI2[sk]] + sB3[sk] * xc[sI3[sk]];
      colS[p * KCP + j] = v;
    }
    __syncthreads();

    // 18 K-steps x 2 WMMA; A fragment shared between the two pos-blocks
#pragma unroll
    for (int kk = 0; kk < KC; kk += 4) {
      const int kb = kk + 2 * hi;  // lanes 0-15: K pair {kk,kk+1}; 16-31: {kk+2,kk+3}
      const v2f a  = *(const v2f*)&wS[ocRow * KCP + kb];
      const v2f b0 = *(const v2f*)&colS[posA * KCP + kb];
      const v2f b1 = *(const v2f*)&colS[posB * KCP + kb];
      acc0 = __builtin_amdgcn_wmma_f32_16x16x4_f32(false, a, false, b0,
                                                   (short)0, acc0, false, false);
      acc1 = __builtin_amdgcn_wmma_f32_16x16x4_f32(false, a, false, b1,
                                                   (short)0, acc1, false, false);
    }
  }

  // ---------- epilogue: bias + store (C/D layout: VGPR v -> M = v + 8*hi) ----------
#pragma unroll
  for (int v = 0; v < 8; ++v) {
    const int oc = (wave & 3) * 16 + v + 8 * hi;
    const float bv = bias[oc];
    float* orow = out + ((size_t)b * OCH + oc) * HW + pos0;
    orow[posA] = acc0[v] + bv;
    orow[posB] = acc1[v] + bv;
  }
}

extern "C" void kernel_launch(void* const* d_in, const int* in_sizes, int n_in,
                              void* d_out, int out_size, void* d_ws, size_t ws_size,
                              hipStream_t stream) {
  const float* x      = (const float*)d_in[0];
  const float* offs   = (const float*)d_in[1];
  const float* weight = (const float*)d_in[2];
  const float* bias   = (const float*)d_in[3];
  float* out          = (float*)d_out;

  dim3 grid(BATCH * NTILES);
  dim3 block(256);
  dcn_fused_wmma_f32<<<grid, block, 0, stream>>>(x, offs, weight, bias, out);
}